// QLayer_29008209117744
// MI455X (gfx1250) — compile-verified
//
#include <hip/hip_runtime.h>
#include <hip/hip_bf16.h>

typedef __attribute__((ext_vector_type(2))) float v2f;
typedef __attribute__((ext_vector_type(8))) float v8f;

#define NEG_SLOPE 0.01f
__device__ __forceinline__ float lrelu(float x) { return x >= 0.f ? x : NEG_SLOPE * x; }

// ---------------------------------------------------------------------------
// Phase 1: for each tile of 16 flat rows (flat = b*2048+n), compute
//   H(16x64) = E(16x64) @ W7^T(64x64)  via V_WMMA_F32_16X16X4_F32,
// then a_i[row] = sum_e lrelu(H[row][e]) * w_i[e],  a_j likewise with w_j.
//
// Layouts (wave32, per ISA 7.12.2):
//   A 16x4 f32 : lane m(<16)/m+16, vgpr v -> A[m][v + 2*half]   => float2 @ off+2*half
//   B 4x16 f32 : lane n(<16)/n+16, vgpr v -> B[v + 2*half][n]   => W7[e][off+2*half..+1]
//   C/D 16x16  : vgpr r, lanes0-15 -> H[r][n], lanes16-31 -> H[r+8][n]
// ---------------------------------------------------------------------------
__global__ __launch_bounds__(32) void aij_kernel(const float* __restrict__ emb,
                                                 const float* __restrict__ W5,
                                                 const float* __restrict__ W7,
                                                 float* __restrict__ ai,
                                                 float* __restrict__ aj) {
  const int lane = threadIdx.x;        // 0..31, EXEC all-ones through the WMMAs
  const int q    = lane & 15;
  const int h    = lane >> 4;
  const int r0   = blockIdx.x << 4;    // flat row base, [0, 16384)

  const float* erow = emb + (size_t)(r0 + q) * 64 + 2 * h;
  const float* b0   = W7  + (size_t)q * 64 + 2 * h;   // e-tile 0 : e = q
  const float* b1   = b0 + 16 * 64;                   // e-tile 1 : e = 16+q
  const float* b2   = b0 + 32 * 64;                   // e-tile 2
  const float* b3   = b0 + 48 * 64;                   // e-tile 3

  v8f c0 = {}, c1 = {}, c2 = {}, c3 = {};
#pragma unroll
  for (int kk = 0; kk < 16; ++kk) {
    const int off = kk * 4;
    v2f a   = *(const v2f*)(erow + off);
    v2f vb0 = *(const v2f*)(b0 + off);
    v2f vb1 = *(const v2f*)(b1 + off);
    v2f vb2 = *(const v2f*)(b2 + off);
    v2f vb3 = *(const v2f*)(b3 + off);
    c0 = __builtin_amdgcn_wmma_f32_16x16x4_f32(false, a, false, vb0, (short)0, c0, false, false);
    c1 = __builtin_amdgcn_wmma_f32_16x16x4_f32(false, a, false, vb1, (short)0, c1, false, false);
    c2 = __builtin_amdgcn_wmma_f32_16x16x4_f32(false, a, false, vb2, (short)0, c2, false, false);
    c3 = __builtin_amdgcn_wmma_f32_16x16x4_f32(false, a, false, vb3, (short)0, c3, false, false);
  }

  // per-lane column weights: this lane owns column e = 16*t + q in tile t
  const float wi0 = W5[64 + q],  wi1 = W5[64 + 16 + q],
              wi2 = W5[64 + 32 + q], wi3 = W5[64 + 48 + q];
  const float wj0 = W5[128 + q], wj1 = W5[128 + 16 + q],
              wj2 = W5[128 + 32 + q], wj3 = W5[128 + 48 + q];

  float p[8], qq[8];
#pragma unroll
  for (int r = 0; r < 8; ++r) {
    float l0 = lrelu(c0[r]), l1 = lrelu(c1[r]), l2 = lrelu(c2[r]), l3 = lrelu(c3[r]);
    p[r]  = l0 * wi0 + l1 * wi1 + l2 * wi2 + l3 * wi3;
    qq[r] = l0 * wj0 + l1 * wj1 + l2 * wj2 + l3 * wj3;
  }
  // reduce across the 16 lanes of each half (mask<16 keeps halves separate)
#pragma unroll
  for (int m = 1; m <= 8; m <<= 1) {
#pragma unroll
    for (int r = 0; r < 8; ++r) {
      p[r]  += __shfl_xor(p[r],  m, 32);
      qq[r] += __shfl_xor(qq[r], m, 32);
    }
  }
  if (q == 0) {                 // lane 0 -> rows r0..r0+7, lane 16 -> rows r0+8..r0+15
    const int rb = r0 + h * 8;
#pragma unroll
    for (int r = 0; r < 8; ++r) { ai[rb + r] = p[r]; aj[rb + r] = qq[r]; }
  }
}

// ---------------------------------------------------------------------------
// Phase 1b: g = sum_n emb ; a_g[b] = lrelu(g@W6^T) . w_g ; noop = g . Wn
// one block per b, 64 threads (one per d / e)
// ---------------------------------------------------------------------------
__global__ __launch_bounds__(64) void g_kernel(const float* __restrict__ emb,
                                               const float* __restrict__ W5,
                                               const float* __restrict__ W6,
                                               const float* __restrict__ Wn,
                                               float* __restrict__ ag,
                                               float* __restrict__ out) {
  const long long NN = 2048LL * 2048LL;
  const int b = blockIdx.x;
  const int d = threadIdx.x;   // 0..63
  __shared__ float gs[64];
  __shared__ float red[64];

  float acc = 0.f;
  const float* p = emb + (size_t)b * 2048 * 64 + d;
  for (int n = 0; n < 2048; ++n) acc += p[(size_t)n * 64];
  gs[d] = acc;
  __syncthreads();

  float gp = 0.f;
  const float* w6row = W6 + (size_t)d * 64;
#pragma unroll 8
  for (int k = 0; k < 64; ++k) gp += gs[k] * w6row[k];
  float ap = lrelu(gp) * W5[d];      // w_g = W5[0:64]
  float np = gs[d] * Wn[d];

  red[d] = ap; __syncthreads();
  for (int s = 32; s > 0; s >>= 1) { if (d < s) red[d] += red[d + s]; __syncthreads(); }
  if (d == 0) ag[b] = red[0];
  __syncthreads();
  red[d] = np; __syncthreads();
  for (int s = 32; s > 0; s >>= 1) { if (d < s) red[d] += red[d + s]; __syncthreads(); }
  if (d == 0) out[(size_t)b * (NN + 1) + NN] = red[0];   // noop_q
}

// ---------------------------------------------------------------------------
// Phase 2: out[b, i*2048 + j] = a_g[b] + a_i[b,i] + a_j[b,j]   (134 MB of stores)
// Row stride NN+1 is odd, so slab base is 16B-aligned only for b%4==0.
// Each slab uses prologue offset kb = (-b) mod 4 so every float4 store aligns.
// Grid: 8 slabs * 4096 blocks * 256 threads, 4 floats/thread.
// ---------------------------------------------------------------------------
__global__ __launch_bounds__(256) void edge_kernel(const float* __restrict__ ai,
                                                   const float* __restrict__ aj,
                                                   const float* __restrict__ ag,
                                                   float* __restrict__ out) {
  const long long NN = 2048LL * 2048LL;
  const int b    = blockIdx.x >> 12;
  const int vblk = blockIdx.x & 4095;
  const int v    = (vblk << 8) + threadIdx.x;        // [0, 1048576)
  const size_t base = (size_t)b * (size_t)(NN + 1);
  const int kb = (4 - (b & 3)) & 3;

  const float s = ag[b];
  const float* aib = ai + (b << 11);
  const float* ajb = aj + (b << 11);

  if (v == 0) {                                       // scalar head [0, kb)
    for (int t = 0; t < kb; ++t)
      out[base + t] = s + aib[t >> 11] + ajb[t & 2047];
  }

  const long long t0 = (long long)kb + 4LL * v;
  if (t0 + 3 < NN) {
    const int j0 = (int)(t0 & 2047);
    float4 r;
    if (j0 <= 2044) {                                 // common case: one row
      const float si = s + aib[(int)(t0 >> 11)];
      r.x = si + ajb[j0];     r.y = si + ajb[j0 + 1];
      r.z = si + ajb[j0 + 2]; r.w = si + ajb[j0 + 3];
    } else {                                          // quad straddles a row edge
      long long t;
      t = t0;     r.x = s + aib[(int)(t >> 11)] + ajb[(int)(t & 2047)];
      t = t0 + 1; r.y = s + aib[(int)(t >> 11)] + ajb[(int)(t & 2047)];
      t = t0 + 2; r.z = s + aib[(int)(t >> 11)] + ajb[(int)(t & 2047)];
      t = t0 + 3; r.w = s + aib[(int)(t >> 11)] + ajb[(int)(t & 2047)];
    }
    *reinterpret_cast<float4*>(out + base + t0) = r;  // 16B-aligned b128 store
  } else {                                            // scalar tail
    for (int c = 0; c < 4; ++c) {
      long long t = t0 + c;
      if (t < NN) out[base + t] = s + aib[(int)(t >> 11)] + ajb[(int)(t & 2047)];
    }
  }
}

extern "C" void kernel_launch(void* const* d_in, const int* in_sizes, int n_in,
                              void* d_out, int out_size, void* d_ws, size_t ws_size,
                              hipStream_t stream) {
  const float* emb = (const float*)d_in[0];   // (8,2048,64)
  const float* W5  = (const float*)d_in[1];   // (1,192)
  const float* W6  = (const float*)d_in[2];   // (64,64)
  const float* W7  = (const float*)d_in[3];   // (64,64)
  const float* Wn  = (const float*)d_in[4];   // (1,64)
  float* out = (float*)d_out;                 // (8, 2048*2048 + 1)

  float* ai = (float*)d_ws;                   // 16384
  float* aj = ai + 16384;                     // 16384
  float* ag = aj + 16384;                     // 8

  aij_kernel<<<1024, 32, 0, stream>>>(emb, W5, W7, ai, aj);
  g_kernel<<<8, 64, 0, stream>>>(emb, W5, W6, Wn, ag, out);
  edge_kernel<<<32768, 256, 0, stream>>>(ai, aj, ag, out);
}